// DeformRouting_6433861009831
// MI455X (gfx1250) — compile-verified
//
#include <hip/hip_runtime.h>
#include <hip/hip_bf16.h>

typedef float v2f __attribute__((ext_vector_type(2)));
typedef float v8f __attribute__((ext_vector_type(8)));

#define NB    4
#define HH    28
#define WW    28
#define CIN   64
#define COUT  64
#define K2    9
#define KDIM  576            // CIN * K2
#define PTOT  3136           // NB * HH * WW  (= 196 * 16, exact)
#define HW    784            // HH * WW
#define NOFF  18             // 2 * K2

// ---------------------------------------------------------------------------
// Kernel 1: offsets[p][j] = b_off[j] + sum_c x[n,c,h,w] * w_off[j,c]
// ---------------------------------------------------------------------------
__global__ void k_offsets(const float* __restrict__ x,
                          const float* __restrict__ w_off,
                          const float* __restrict__ b_off,
                          float* __restrict__ off_ws) {
  int tid = blockIdx.x * blockDim.x + threadIdx.x;
  if (tid >= PTOT * NOFF) return;
  int p = tid / NOFF;
  int j = tid - p * NOFF;
  int n = p / HW;
  int hw = p - n * HW;           // h*28 + w
  const float* xp = x + n * (CIN * HW) + hw;
  const float* wr = w_off + j * CIN;
  float acc = b_off[j];
#pragma unroll 8
  for (int c = 0; c < CIN; ++c)
    acc += xp[c * HW] * wr[c];
  off_ws[p * NOFF + j] = acc;
}

// ---------------------------------------------------------------------------
// Kernel 2: bilinear deformable sampling -> patches[p][c*9 + ky*3 + kx]
// One block per pixel, one thread per input channel.
// ---------------------------------------------------------------------------
__global__ void k_patches(const float* __restrict__ x,
                          const float* __restrict__ off_ws,
                          float* __restrict__ patches) {
  int p = blockIdx.x;            // 0 .. PTOT-1
  int c = threadIdx.x;           // 0 .. 63
  int n  = p / HW;
  int hw = p - n * HW;
  int h  = hw / WW;
  int w  = hw - h * WW;

  const float inv27 = 1.0f / 27.0f;
  float xs = 2.0f * (float)w * inv27 - 1.0f;   // linspace(-1,1,28)[w]
  float ys = 2.0f * (float)h * inv27 - 1.0f;
  float kstep = 2.0f * inv27;                  // (k-1)/(dim-1)*DIL spacing

  const float* xc0 = x + (n * CIN + c) * HW;   // channel plane
  const float* offp = off_ws + p * NOFF;
  float* outp = patches + (size_t)p * KDIM + c * K2;

#pragma unroll
  for (int ty = 0; ty < 3; ++ty) {
#pragma unroll
    for (int tx = 0; tx < 3; ++tx) {
      float gx = offp[ty * 6 + tx * 2 + 0] + xs + (float)(tx - 1) * kstep;
      float gy = offp[ty * 6 + tx * 2 + 1] + ys + (float)(ty - 1) * kstep;
      float ix = (gx + 1.0f) * 13.5f;          // 0.5*(W-1)
      float iy = (gy + 1.0f) * 13.5f;
      float ix0 = floorf(ix);
      float iy0 = floorf(iy);
      float fx = ix - ix0;
      float fy = iy - iy0;
      float v = 0.0f;
#pragma unroll
      for (int dy = 0; dy < 2; ++dy) {
#pragma unroll
        for (int dx = 0; dx < 2; ++dx) {
          float xx = ix0 + (float)dx;
          float yy = iy0 + (float)dy;
          bool valid = (xx >= 0.0f) && (xx <= 27.0f) && (yy >= 0.0f) && (yy <= 27.0f);
          int xi = (int)fminf(fmaxf(xx, 0.0f), 27.0f);
          int yi = (int)fminf(fmaxf(yy, 0.0f), 27.0f);
          float s = valid ? xc0[yi * WW + xi] : 0.0f;
          float wgt = (dy ? fy : (1.0f - fy)) * (dx ? fx : (1.0f - fx));
          v += s * wgt;
        }
      }
      outp[ty * 3 + tx] = v;
    }
  }
}

// ---------------------------------------------------------------------------
// Kernel 3: WMMA fp32 GEMM epilogue-fused.
//   accW[p,o] = Patches(p,:) . w_wgt(o,:)   (K = 576)
//   accB[p,o] = Patches(p,:) . b_wgt(o,:)
//   out[n,o,h,w] = x[n,o,h,w] * accW + accB
// One wave per 16(M pixels) x 16(N outch) tile; 196*4 = 784 tiles.
// ---------------------------------------------------------------------------
#define TILES_M 196
#define TILES_N 4
#define NTILES  (TILES_M * TILES_N)

__global__ void k_gemm(const float* __restrict__ patches,
                       const float* __restrict__ w_wgt,
                       const float* __restrict__ b_wgt,
                       const float* __restrict__ x,
                       float* __restrict__ out) {
  int wave = (blockIdx.x * blockDim.x + threadIdx.x) >> 5;
  if (wave >= NTILES) return;
  int lane = threadIdx.x & 31;
  int half = lane >> 4;          // 0: K pair {0,1}; 1: K pair {2,3}
  int col  = lane & 15;

  int nTile = wave & 3;          // output-channel tile
  int mTile = wave >> 2;         // pixel tile
  int mBase = mTile * 16;
  int oBase = nTile * 16;

  const float* aRow  = patches + (size_t)(mBase + col) * KDIM + 2 * half;
  const float* bRowW = w_wgt   + (size_t)(oBase + col) * KDIM + 2 * half;
  const float* bRowB = b_wgt   + (size_t)(oBase + col) * KDIM + 2 * half;

  v8f accW = {};
  v8f accB = {};

#pragma unroll 4
  for (int k0 = 0; k0 < KDIM; k0 += 4) {
    v2f a  = *(const v2f*)(aRow  + k0);
    v2f bw = *(const v2f*)(bRowW + k0);
    v2f bb = *(const v2f*)(bRowB + k0);
    accW = __builtin_amdgcn_wmma_f32_16x16x4_f32(
        false, a, false, bw, (short)0, accW, false, false);
    accB = __builtin_amdgcn_wmma_f32_16x16x4_f32(
        false, a, false, bb, (short)0, accB, false, false);
  }

  // Epilogue: D layout -> VGPR r holds (M = r + 8*half, N = col)
  int o = oBase + col;
#pragma unroll
  for (int r = 0; r < 8; ++r) {
    int p = mBase + r + 8 * half;
    int n = p / HW;
    int hw = p - n * HW;
    size_t idx = (size_t)(n * COUT + o) * HW + hw;
    out[idx] = x[idx] * accW[r] + accB[r];
  }
}

// ---------------------------------------------------------------------------
extern "C" void kernel_launch(void* const* d_in, const int* in_sizes, int n_in,
                              void* d_out, int out_size, void* d_ws, size_t ws_size,
                              hipStream_t stream) {
  const float* x     = (const float*)d_in[0];
  const float* w_off = (const float*)d_in[1];
  const float* b_off = (const float*)d_in[2];
  const float* w_wgt = (const float*)d_in[3];
  const float* b_wgt = (const float*)d_in[4];
  float* out = (float*)d_out;

  // Workspace layout: [offsets: PTOT*18 floats][patches: PTOT*576 floats]
  float* off_ws  = (float*)d_ws;
  float* patches = off_ws + 57344;   // 56448 rounded up, keeps 256B alignment

  {
    int threads = PTOT * NOFF;                 // 56448
    int blk = 256;
    k_offsets<<<(threads + blk - 1) / blk, blk, 0, stream>>>(x, w_off, b_off, off_ws);
  }
  {
    k_patches<<<PTOT, CIN, 0, stream>>>(x, off_ws, patches);
  }
  {
    int blk = 256;                             // 8 waves per block
    int waves_per_blk = blk / 32;
    int blocks = (NTILES + waves_per_blk - 1) / waves_per_blk;  // 98
    k_gemm<<<blocks, blk, 0, stream>>>(patches, w_wgt, b_wgt, x, out);
  }
}